// PlanNotesProjection_31610959298857
// MI455X (gfx1250) — compile-verified
//
#include <hip/hip_runtime.h>

// Problem constants (from reference setup_inputs)
#define B_  8
#define K_  32
#define S_  4096
#define H_  2048
#define D_  1024
#define SCH 4                    // S split into 4 chunks for wave parallelism
#define SCHLEN (S_ / SCH)        // 1024

typedef float v2f __attribute__((ext_vector_type(2)));
typedef float v8f __attribute__((ext_vector_type(8)));

// ---------------------------------------------------------------------------
// Kernel 0: per-(b,k) ownership counts -> 1 / max(count, 1)
// ---------------------------------------------------------------------------
__global__ __launch_bounds__(256)
void counts_kernel(const unsigned char* __restrict__ own,
                   float* __restrict__ invcnt) {
  __shared__ float red[256];
  const int r = blockIdx.x;                 // 0..255 = b*K + k
  const int t = threadIdx.x;
  const unsigned char* p = own + (size_t)r * S_;
  float s = 0.f;
  for (int i = t; i < S_; i += 256) s += (float)p[i];
  red[t] = s;
  __syncthreads();
  for (int off = 128; off > 0; off >>= 1) {
    if (t < off) red[t] += red[t + off];
    __syncthreads();
  }
  if (t == 0) invcnt[r] = 1.0f / fmaxf(red[0], 1.0f);
}

// ---------------------------------------------------------------------------
// Kernel 1: ownership-weighted pooling as fp32 WMMA GEMM.
//   C[b, 0..31, n0..n0+15] += own[b, m, s] * plan[b, s, n]
// One wave = one 16-wide N tile, BOTH 16-row M tiles (reuses B fragment),
// one S chunk. Partials go to parts[chunk][256][H].
// ---------------------------------------------------------------------------
__global__ __launch_bounds__(256)
void pool_kernel(const float* __restrict__ plan,
                 const unsigned char* __restrict__ own,
                 float* __restrict__ parts) {
  const int lane = threadIdx.x & 31;
  const int wave = threadIdx.x >> 5;            // 0..7
  const int n0   = (blockIdx.x * 8 + wave) * 16; // N tile base in H
  const int chunk = blockIdx.y;                  // 0..SCH-1
  const int b    = blockIdx.z;
  const int m    = lane & 15;
  const int half = lane >> 4;                    // 0 -> K 0,1 ; 1 -> K 2,3
  const int s0   = chunk * SCHLEN;

  // A rows for both M tiles (ownership bytes), pre-offset by lane's K pair
  const unsigned char* arow0 =
      own + ((size_t)(b * K_) + m) * S_ + s0 + half * 2;
  const unsigned char* arow1 = arow0 + (size_t)16 * S_;
  // B column for this lane: plan[b, k, n0+m]
  const float* bcol =
      plan + (size_t)b * S_ * H_ + (size_t)(s0 + half * 2) * H_ + (n0 + m);

  v8f acc0 = {};
  v8f acc1 = {};
#pragma unroll 4
  for (int s = 0; s < SCHLEN; s += 4) {
    v2f a0, a1, bb;
    a0.x = (float)arow0[s];
    a0.y = (float)arow0[s + 1];
    a1.x = (float)arow1[s];
    a1.y = (float)arow1[s + 1];
    bb.x = bcol[(size_t)s * H_];
    bb.y = bcol[(size_t)s * H_ + H_];
    acc0 = __builtin_amdgcn_wmma_f32_16x16x4_f32(
        false, a0, false, bb, (short)0, acc0, false, false);
    acc1 = __builtin_amdgcn_wmma_f32_16x16x4_f32(
        false, a1, false, bb, (short)0, acc1, false, false);
  }

  // C/D layout: VGPR i -> row i (lanes 0-15) / row i+8 (lanes 16-31), col=lane%16
  const size_t pbase = (size_t)chunk * (B_ * K_) * H_;
  const int kbase = b * K_;
#pragma unroll
  for (int i = 0; i < 8; ++i) {
    const int row = i + half * 8;
    parts[pbase + (size_t)(kbase + row) * H_ + n0 + m] = acc0[i];
    parts[pbase + (size_t)(kbase + 16 + row) * H_ + n0 + m] = acc1[i];
  }
}

// ---------------------------------------------------------------------------
// Kernel 2: fold S-chunk partials and scale by 1/count -> pooled[256][H]
// ---------------------------------------------------------------------------
__global__ __launch_bounds__(256)
void reduce_kernel(const float* __restrict__ parts,
                   const float* __restrict__ invcnt,
                   float* __restrict__ pooled) {
  const size_t idx = (size_t)blockIdx.x * 256 + threadIdx.x; // over 256*H
  const int row = (int)(idx >> 11);                          // H_ == 2048
  float s = 0.f;
#pragma unroll
  for (int c = 0; c < SCH; ++c)
    s += parts[(size_t)c * (B_ * K_) * H_ + idx];
  pooled[idx] = s * invcnt[row];
}

// ---------------------------------------------------------------------------
// Kernel 3: projection GEMM (pooled @ W + b) + LayerNorm over D.
// One block = 16 output rows; 16 waves x 4 N-tiles; fp32 WMMA K-loop,
// result staged in LDS, one wave per row does the LN reduction (wave32
// __shfl_xor butterflies).
// ---------------------------------------------------------------------------
__global__ __launch_bounds__(512)
void proj_ln_kernel(const float* __restrict__ pooled,
                    const float* __restrict__ Wm,
                    const float* __restrict__ bias,
                    const float* __restrict__ gamma,
                    const float* __restrict__ beta,
                    float* __restrict__ out) {
  __shared__ float tile[16 * D_];              // 64 KB of the 320 KB LDS
  const int lane = threadIdx.x & 31;
  const int wave = threadIdx.x >> 5;           // 0..15
  const int mt   = blockIdx.x;                 // 16-row M tile (rows mt*16..)
  const int m    = lane & 15;
  const int half = lane >> 4;
  const int n0   = wave * 16;                  // tiles at n0 + j*256

  const float* arow = pooled + ((size_t)(mt * 16) + m) * H_ + half * 2;
  const float* wbase = Wm + (size_t)(half * 2) * D_ + m;

  v8f acc[4] = {v8f{}, v8f{}, v8f{}, v8f{}};
#pragma unroll 2
  for (int k = 0; k < H_; k += 4) {
    v2f a;
    a.x = arow[k];
    a.y = arow[k + 1];
    const size_t wo = (size_t)k * D_;
#pragma unroll
    for (int j = 0; j < 4; ++j) {
      v2f bb;
      bb.x = wbase[wo + n0 + j * 256];
      bb.y = wbase[wo + D_ + n0 + j * 256];
      acc[j] = __builtin_amdgcn_wmma_f32_16x16x4_f32(
          false, a, false, bb, (short)0, acc[j], false, false);
    }
  }

  // Stage projected tile (+bias) into LDS
#pragma unroll
  for (int j = 0; j < 4; ++j) {
    const int col = n0 + j * 256 + m;
#pragma unroll
    for (int i = 0; i < 8; ++i) {
      const int row = i + half * 8;
      tile[row * D_ + col] = acc[j][i] + bias[col];
    }
  }
  __syncthreads();

  // LayerNorm: wave w owns row w (wave32 -> 32 lanes x 32 cols each)
  const int row = wave;
  float s1 = 0.f, s2 = 0.f;
#pragma unroll
  for (int c = lane; c < D_; c += 32) {
    const float x = tile[row * D_ + c];
    s1 += x;
    s2 += x * x;
  }
#pragma unroll
  for (int off = 16; off > 0; off >>= 1) {
    s1 += __shfl_xor(s1, off, 32);
    s2 += __shfl_xor(s2, off, 32);
  }
  const float mu   = s1 * (1.0f / D_);
  const float var  = fmaxf(s2 * (1.0f / D_) - mu * mu, 0.0f);
  const float rstd = rsqrtf(var + 1e-5f);

  const size_t obase = (size_t)(mt * 16 + row) * D_;
#pragma unroll
  for (int c = lane; c < D_; c += 32) {
    const float x = (tile[row * D_ + c] - mu) * rstd;
    out[obase + c] = x * gamma[c] + beta[c];
  }
}

// ---------------------------------------------------------------------------
extern "C" void kernel_launch(void* const* d_in, const int* in_sizes, int n_in,
                              void* d_out, int out_size, void* d_ws, size_t ws_size,
                              hipStream_t stream) {
  (void)in_sizes; (void)n_in; (void)out_size; (void)ws_size;
  const float*         plan  = (const float*)d_in[0];          // (B,S,H) f32
  const unsigned char* own   = (const unsigned char*)d_in[1];  // (B,K,S) bool
  const float*         Wm    = (const float*)d_in[2];          // (H,D) f32
  const float*         bias  = (const float*)d_in[3];          // (D,)
  const float*         gamma = (const float*)d_in[4];          // (D,)
  const float*         beta  = (const float*)d_in[5];          // (D,)
  float*               out   = (float*)d_out;                  // (B,K,D) f32

  // Workspace layout: invcnt[256] | pooled[256*H] | parts[SCH*256*H]
  float* invcnt = (float*)d_ws;
  float* pooled = invcnt + 256;
  float* parts  = pooled + (size_t)(B_ * K_) * H_;

  counts_kernel<<<dim3(B_ * K_), dim3(256), 0, stream>>>(own, invcnt);

  pool_kernel<<<dim3(H_ / 16 / 8, SCH, B_), dim3(256), 0, stream>>>(
      plan, own, parts);

  reduce_kernel<<<dim3((B_ * K_) * H_ / 256), dim3(256), 0, stream>>>(
      parts, invcnt, pooled);

  proj_ln_kernel<<<dim3((B_ * K_) / 16), dim3(512), 0, stream>>>(
      pooled, Wm, bias, gamma, beta, out);
}